// AutoCorrelationLayer_377957122232
// MI455X (gfx1250) — compile-verified
//
#include <hip/hip_runtime.h>
#include <cstdint>

// Problem constants (fixed by the reference): B=32, L=4096, H=8, D=64, top-k = 5*ceil(ln 4096) = 45
#define BATCH 32
#define LSEQ  4096
#define HEADS 8
#define DHEAD 64
#define NBH   (BATCH*HEADS)   // 256
#define TOPK  45

typedef __attribute__((ext_vector_type(2))) float v2f;
typedef __attribute__((ext_vector_type(8))) float v8f;

// -----------------------------------------------------------------------------
// Kernel 1: fused q/k/v projection.  row = (b*L + l)*H + h ; out[(b*H+h)*L + l]
// 8 lanes per row, each lane loads 2 float4 (32B) -> wave reads 1KB contiguous.
// Memory bound: 805 MB of reads (~35us at 23.3 TB/s) dominates the whole op.
// -----------------------------------------------------------------------------
__global__ __launch_bounds__(256) void proj_kernel(
    const float* __restrict__ Qin, const float* __restrict__ Kin, const float* __restrict__ Vin,
    const float* __restrict__ Wq, const float* __restrict__ bq,
    const float* __restrict__ Wk, const float* __restrict__ bk,
    const float* __restrict__ Wv, const float* __restrict__ bv,
    float* __restrict__ qo, float* __restrict__ ko, float* __restrict__ vo)
{
  const int tid = threadIdx.x;
  const int sub = tid & 7;                       // lane within 8-lane row group
  const int row = blockIdx.x * 32 + (tid >> 3);  // (b*L + l)*H + h
  const int h = row & (HEADS - 1);
  const int l = (row >> 3) & (LSEQ - 1);
  const int b = row >> 15;
  const int oidx = (b << 15) + (h << 12) + l;    // (b*H+h)*L + l
  const size_t base = (size_t)row * DHEAD + sub * 8;

  const float* ins[3]  = {Qin, Kin, Vin};
  const float* Ws[3]   = {Wq, Wk, Wv};
  const float* bs[3]   = {bq, bk, bv};
  float*       outs[3] = {qo, ko, vo};

  #pragma unroll
  for (int t = 0; t < 3; ++t) {
    const float4* src4 = (const float4*)(ins[t] + base);
    float4 x0 = src4[0];
    float4 x1 = src4[1];
    const float4* W4 = (const float4*)Ws[t];
    float4 w0 = W4[sub * 2];
    float4 w1 = W4[sub * 2 + 1];
    float p = x0.x*w0.x + x0.y*w0.y + x0.z*w0.z + x0.w*w0.w
            + x1.x*w1.x + x1.y*w1.y + x1.z*w1.z + x1.w*w1.w;
    p += __shfl_xor(p, 4, 8);
    p += __shfl_xor(p, 2, 8);
    p += __shfl_xor(p, 1, 8);
    if (sub == 0) outs[t][oidx] = p + bs[t][0];
  }
}

// -----------------------------------------------------------------------------
// Kernel 2: exact fp32 autocorrelation on the matrix core.
//   score[tau] = (1/L) * sum_{t>=tau} q[t]*k[t-tau]
// GEMM form:  D[i,j] = score[tau0 + 16i + j] = sum_c A[i,c]*B[c,j]
//   A[i,c] = q[tau0 + c + 16i] (Toeplitz),  B[c,j] = k[c - j]
// V_WMMA_F32_16X16X4_F32 chain; scalar (uniform) trip count so the loop is a
// pure s_cbranch loop with EXEC all-1s; unrolled x4 with 2 accumulators so the
// body is 8x ds_load_2addr + 4x wmma with overlapped waits.
// qs swizzle (+1 word per 16) makes the stride-16 A gather bank-conflict-free;
// the swizzled address advances by exactly 34 per 32-element K step.
// -----------------------------------------------------------------------------
__global__ __launch_bounds__(256) void corr_kernel(
    const float* __restrict__ q, const float* __restrict__ k, float* __restrict__ score)
{
  __shared__ float qs[4624];   // (4096+256) + swizzle pad
  __shared__ float ks[4128];   // (4096+32)
  __shared__ float red[2048];  // 8 waves x 256 partial-tile values

  const int tid  = threadIdx.x;
  const int st   = blockIdx.x;        // supertile 0..15
  const int bh   = blockIdx.y;        // 0..255
  const int tau0 = st << 8;
  const int n    = LSEQ - tau0;       // reduction length; multiple of 256
  const float* qg = q + (size_t)bh * LSEQ;
  const float* kg = k + (size_t)bh * LSEQ;

  // Stage q[tau0 + x] (zero-padded above) with swizzle x -> x + x/16
  for (int x = tid; x < n + 256; x += 256) {
    float vq = (x < n) ? qg[tau0 + x] : 0.f;
    qs[x + (x >> 4)] = vq;
  }
  // Stage k[x-16] (zero-padded below and above): B[c,j] = ks[c - j + 16]
  for (int x = tid; x < n + 32; x += 256) {
    float vk = (x >= 16 && x < n + 16) ? kg[x - 16] : 0.f;
    ks[x] = vk;
  }
  __syncthreads();

  const int lane = tid & 31;
  const int wave = tid >> 5;
  const int m    = lane & 15;   // A row / B col
  const int half = lane >> 4;   // K pair select (K = 2*half, 2*half+1)

  v8f acc0 = {0.f,0.f,0.f,0.f,0.f,0.f,0.f,0.f};
  v8f acc1 = {0.f,0.f,0.f,0.f,0.f,0.f,0.f,0.f};

  // Per-lane starting addresses; trip count is uniform: every wave runs n/32
  // iterations (c = wave*4 + 32*it, all < n since n is a multiple of 32).
  int qa   = wave * 4 + 16 * m + 2 * half;   // logical q index (even)
  int qidx = qa + (qa >> 4);                 // swizzled; advances by 34 per step
  int kidx = wave * 4 + 2 * half - m + 16;   // advances by 32 per step
  const int trips = n >> 5;                  // multiple of 8 -> unroll 4 is exact

  for (int it = 0; it < trips; it += 4) {
    v2f a0, a1, a2, a3, b0, b1, b2, b3;
    a0.x = qs[qidx];       a0.y = qs[qidx + 1];
    a1.x = qs[qidx + 34];  a1.y = qs[qidx + 35];
    a2.x = qs[qidx + 68];  a2.y = qs[qidx + 69];
    a3.x = qs[qidx + 102]; a3.y = qs[qidx + 103];
    b0.x = ks[kidx];       b0.y = ks[kidx + 1];
    b1.x = ks[kidx + 32];  b1.y = ks[kidx + 33];
    b2.x = ks[kidx + 64];  b2.y = ks[kidx + 65];
    b3.x = ks[kidx + 96];  b3.y = ks[kidx + 97];
    acc0 = __builtin_amdgcn_wmma_f32_16x16x4_f32(false, a0, false, b0, (short)0, acc0, false, false);
    acc1 = __builtin_amdgcn_wmma_f32_16x16x4_f32(false, a1, false, b1, (short)0, acc1, false, false);
    acc0 = __builtin_amdgcn_wmma_f32_16x16x4_f32(false, a2, false, b2, (short)0, acc0, false, false);
    acc1 = __builtin_amdgcn_wmma_f32_16x16x4_f32(false, a3, false, b3, (short)0, acc1, false, false);
    qidx += 136;   // 4 * 34
    kidx += 128;   // 4 * 32
  }

  // Reduce the 8 partial 16x16 tiles across waves via LDS.
  #pragma unroll
  for (int e = 0; e < 8; ++e) red[wave * 256 + lane * 8 + e] = acc0[e] + acc1[e];
  __syncthreads();

  float s = 0.f;
  #pragma unroll
  for (int w = 0; w < 8; ++w) s += red[w * 256 + tid];

  // C/D layout: VGPR e, lane l -> M = e + 8*(l>>4), N = l&15 ; tau = tau0 + 16*M + N
  const int rl  = tid >> 3;   // lane of the owning slot
  const int re  = tid & 7;    // VGPR index
  const int tau = tau0 + 16 * (re + 8 * (rl >> 4)) + (rl & 15);
  score[(size_t)bh * LSEQ + tau] = s * (1.f / (float)LSEQ);
}

// -----------------------------------------------------------------------------
// Kernel 3: per-head top-45 (ties -> lower index like lax.top_k), softmax,
// then out[l] = sum_j w_j * v[(idx_j + l) & 4095].
// Scores live in registers (16/thread); argmax = register scan + wave shuffle
// reduce + one LDS hop for 8 wave winners -> only 2 barriers per iteration.
// -----------------------------------------------------------------------------
__global__ __launch_bounds__(256) void topk_out_kernel(
    const float* __restrict__ score, const float* __restrict__ v, float* __restrict__ out)
{
  __shared__ float vv[LSEQ];
  __shared__ float rmax[8];
  __shared__ int   ridx[8];
  __shared__ float wts[TOPK];
  __shared__ int   idxs[TOPK];

  const int tid  = threadIdx.x;
  const int lane = tid & 31;
  const int wave = tid >> 5;
  const int bh   = blockIdx.x;
  const float* sg = score + (size_t)bh * LSEQ;
  const float* vg = v + (size_t)bh * LSEQ;

  float loc[16];                       // thread owns scores i = r*256 + tid
  #pragma unroll
  for (int r = 0; r < 16; ++r) loc[r] = sg[r * 256 + tid];
  for (int i = tid; i < LSEQ; i += 256) vv[i] = vg[i];
  __syncthreads();

  for (int j = 0; j < TOPK; ++j) {
    // register argmax (ascending r + strict '>' keeps the lowest index)
    float m = loc[0]; int mi = tid;
    #pragma unroll
    for (int r = 1; r < 16; ++r) {
      float vr = loc[r]; int ir = r * 256 + tid;
      if (vr > m) { m = vr; mi = ir; }
    }
    // wave32 shuffle reduce
    #pragma unroll
    for (int off = 16; off > 0; off >>= 1) {
      float om = __shfl_xor(m, off);
      int   oi = __shfl_xor(mi, off);
      if (om > m || (om == m && oi < mi)) { m = om; mi = oi; }
    }
    if (lane == 0) { rmax[wave] = m; ridx[wave] = mi; }
    __syncthreads();
    if (wave == 0) {
      float m2  = (lane < 8) ? rmax[lane] : -3.4e38f;
      int   mi2 = (lane < 8) ? ridx[lane] : 0x7fffffff;
      #pragma unroll
      for (int off = 4; off > 0; off >>= 1) {
        float om = __shfl_xor(m2, off, 8);
        int   oi = __shfl_xor(mi2, off, 8);
        if (om > m2 || (om == m2 && oi < mi2)) { m2 = om; mi2 = oi; }
      }
      if (lane == 0) { idxs[j] = mi2; wts[j] = m2; }
    }
    __syncthreads();
    // knock the winner out of its owner's registers (no dynamic reg indexing)
    const int sel = idxs[j];
    if ((sel & 255) == tid) {
      const int rr = sel >> 8;
      #pragma unroll
      for (int r = 0; r < 16; ++r) if (r == rr) loc[r] = -3.4e38f;
    }
  }

  if (tid == 0) {
    float mx = wts[0];
    for (int j = 1; j < TOPK; ++j) mx = fmaxf(mx, wts[j]);
    float ssum = 0.f;
    for (int j = 0; j < TOPK; ++j) { float e = __expf(wts[j] - mx); wts[j] = e; ssum += e; }
    float inv = 1.f / ssum;
    for (int j = 0; j < TOPK; ++j) wts[j] *= inv;
  }
  __syncthreads();

  for (int l2 = tid; l2 < LSEQ; l2 += 256) {
    float acc = 0.f;
    #pragma unroll 5
    for (int j = 0; j < TOPK; ++j) acc += wts[j] * vv[(idxs[j] + l2) & (LSEQ - 1)];
    out[(size_t)bh * LSEQ + l2] = acc;
  }
}

// -----------------------------------------------------------------------------
extern "C" void kernel_launch(void* const* d_in, const int* in_sizes, int n_in,
                              void* d_out, int out_size, void* d_ws, size_t ws_size,
                              hipStream_t stream) {
  const float* Qin = (const float*)d_in[0];
  const float* Kin = (const float*)d_in[1];
  const float* Vin = (const float*)d_in[2];
  const float* Wq  = (const float*)d_in[3];
  const float* bq  = (const float*)d_in[4];
  const float* Wk  = (const float*)d_in[5];
  const float* bk  = (const float*)d_in[6];
  const float* Wv  = (const float*)d_in[7];
  const float* bv  = (const float*)d_in[8];
  float* out = (float*)d_out;

  // Workspace: q, k, v, score -> 4 * (256*4096) floats = 16 MB
  float* q     = (float*)d_ws;
  float* k     = q + (size_t)NBH * LSEQ;
  float* v     = k + (size_t)NBH * LSEQ;
  float* score = v + (size_t)NBH * LSEQ;

  // 1) projection: 1,048,576 rows, 32 rows/block
  proj_kernel<<<dim3((BATCH * LSEQ * HEADS) / 32), dim3(256), 0, stream>>>(
      Qin, Kin, Vin, Wq, bq, Wk, bk, Wv, bv, q, k, v);

  // 2) WMMA autocorrelation: 16 supertiles x 256 heads
  corr_kernel<<<dim3(16, NBH), dim3(256), 0, stream>>>(q, k, score);

  // 3) top-k + softmax + aggregation: one block per head
  topk_out_kernel<<<dim3(NBH), dim3(256), 0, stream>>>(score, v, out);
}